// LibraKANBlock_56083682951291
// MI455X (gfx1250) — compile-verified
//
#include <hip/hip_runtime.h>
#include <hip/hip_bf16.h>

#define WIDTH 512
#define NTOK 4096
#define FTOT 1024
#define RHO 0.25f
#define SPECTRAL_SCALE 0.8f
#define ES_BETA 6.0f
#define TAU 0.001f
#define EPS_RMS 1e-8f
#define LN_EPS 1e-5f

typedef __attribute__((ext_vector_type(2))) float v2f;
typedef __attribute__((ext_vector_type(8))) float v8f;

// ---------------------------------------------------------------------------
// 1) init + abs-max reduction over freq (for the spectral window denominator)
// ---------------------------------------------------------------------------
__global__ void absmax_init_kernel(unsigned* p) { *p = 0u; }

__global__ __launch_bounds__(256) void absmax_kernel(const float* __restrict__ freq,
                                                     unsigned* __restrict__ amax, int n) {
    float m = 0.0f;
    for (int i = blockIdx.x * blockDim.x + threadIdx.x; i < n; i += gridDim.x * blockDim.x)
        m = fmaxf(m, fabsf(freq[i]));
    #pragma unroll
    for (int o = 16; o > 0; o >>= 1) m = fmaxf(m, __shfl_xor(m, o, 32));
    __shared__ float red[8];
    const int lane = threadIdx.x & 31, wave = threadIdx.x >> 5;
    if (lane == 0) red[wave] = m;
    __syncthreads();
    if (threadIdx.x == 0) {
        #pragma unroll
        for (int j = 1; j < 8; ++j) m = fmaxf(m, red[j]);
        atomicMax(amax, __float_as_uint(m));  // m >= 0: uint order == float order
    }
}

// ---------------------------------------------------------------------------
// 2) LayerNorm: one wave per token row (32 lanes x 16 elems), float4 coalesced
// ---------------------------------------------------------------------------
__global__ __launch_bounds__(256) void ln_kernel(const float* __restrict__ x,
                                                 const float* __restrict__ gamma,
                                                 const float* __restrict__ beta,
                                                 float* __restrict__ xn) {
    const int lane = threadIdx.x & 31;
    const int wave = threadIdx.x >> 5;
    const int row  = blockIdx.x * 8 + wave;
    const float* xr = x + (size_t)row * WIDTH;

    float4 v[4];
    float s = 0.0f, ss = 0.0f;
    #pragma unroll
    for (int j = 0; j < 4; ++j) {
        v[j] = *(const float4*)(xr + (lane + 32 * j) * 4);
        s  += v[j].x + v[j].y + v[j].z + v[j].w;
        ss += v[j].x * v[j].x + v[j].y * v[j].y + v[j].z * v[j].z + v[j].w * v[j].w;
    }
    #pragma unroll
    for (int o = 16; o > 0; o >>= 1) {
        s  += __shfl_xor(s, o, 32);
        ss += __shfl_xor(ss, o, 32);
    }
    const float mu  = s  * (1.0f / WIDTH);
    const float var = ss * (1.0f / WIDTH) - mu * mu;
    const float inv = rsqrtf(var + LN_EPS);

    float* xo = xn + (size_t)row * WIDTH;
    #pragma unroll
    for (int j = 0; j < 4; ++j) {
        const int e = (lane + 32 * j) * 4;
        const float4 g = *(const float4*)(gamma + e);
        const float4 b = *(const float4*)(beta + e);
        float4 o;
        o.x = (v[j].x - mu) * inv * g.x + b.x;
        o.y = (v[j].y - mu) * inv * g.y + b.y;
        o.z = (v[j].z - mu) * inv * g.z + b.z;
        o.w = (v[j].w - mu) * inv * g.w + b.w;
        *(float4*)(xo + e) = o;
    }
}

// ---------------------------------------------------------------------------
// 3) fused spectral coefficient: thr[c,w] = sign(a*g)*relu(|a*g|-tau)*win
// ---------------------------------------------------------------------------
__global__ __launch_bounds__(256) void thr_kernel(const float* __restrict__ freq,
                                                  const float* __restrict__ alpha,
                                                  const float* __restrict__ gate,
                                                  const unsigned* __restrict__ amax,
                                                  float* __restrict__ thr) {
    const int i = blockIdx.x * blockDim.x + threadIdx.x;
    const float fm    = __uint_as_float(*amax);       // max |freq|
    const float denom = fm * SPECTRAL_SCALE;          // max |0.8*freq|
    const float fr    = freq[i];
    float win;
    if (denom < 1e-8f) {
        win = 1.0f;
    } else {
        const float r = fr / fm;                      // (0.8 f)/(0.8 fmax)
        win = __expf(-ES_BETA * r * r);
    }
    const float raw = alpha[i] * gate[i];
    const float aw  = fabsf(raw) - TAU;
    const float t   = (aw > 0.0f) ? copysignf(aw, raw) : 0.0f;
    thr[i] = t * win;
}

// ---------------------------------------------------------------------------
// 4) Local branch GEMM via V_WMMA_F32_16X16X4_F32 + bias + exact-erf GELU.
//    Block = 256 threads = 8 waves; block tile 32(M) x 64(N); K-chunk 32 in LDS.
//    LDS row stride 36: float4 stores stay 16B-aligned, and the 16x16x4 f32
//    lane layout (k = kk + 2*hi + vgpr) reads conflict-free ds_load_b64 pairs.
// ---------------------------------------------------------------------------
__global__ __launch_bounds__(256) void gemm_gelu_kernel(const float* __restrict__ xn,
                                                        const float* __restrict__ w,
                                                        const float* __restrict__ bias,
                                                        float* __restrict__ out) {
    __shared__ __align__(16) float As[32][36];
    __shared__ __align__(16) float Bs[64][36];

    const int tid  = threadIdx.x;
    const int lane = tid & 31;
    const int wave = tid >> 5;
    const int l16  = lane & 15;
    const int hi   = lane >> 4;
    const int m0   = blockIdx.x * 32;
    const int n0   = blockIdx.y * 64;
    const int wm   = (wave & 1) * 16;   // wave tile row within block
    const int wn   = (wave >> 1) * 16;  // wave tile col within block

    // cooperative loader coordinates: 256 threads, float4 each
    const int lr = tid >> 3;            // 0..31
    const int lc = (tid & 7) * 4;       // 0..28

    v8f acc = {};
    for (int k0 = 0; k0 < WIDTH; k0 += 32) {
        // A: x_n tile [32 x 32]
        *(float4*)(&As[lr][lc]) =
            *(const float4*)(xn + (size_t)(m0 + lr) * WIDTH + k0 + lc);
        // B: w_loc tile [64 x 32]  (B[k][n] = w_loc[n][k], stored [n][k])
        *(float4*)(&Bs[lr][lc]) =
            *(const float4*)(w + (size_t)(n0 + lr) * WIDTH + k0 + lc);
        *(float4*)(&Bs[lr + 32][lc]) =
            *(const float4*)(w + (size_t)(n0 + lr + 32) * WIDTH + k0 + lc);
        __syncthreads();

        #pragma unroll
        for (int kk = 0; kk < 32; kk += 4) {
            // A lane layout: m = lane%16, k = kk + 2*hi + {0,1}  -> one b64
            v2f a = *(const v2f*)(&As[wm + l16][kk + 2 * hi]);
            // B lane layout: n = lane%16, k = kk + 2*hi + {0,1}  -> one b64
            v2f b = *(const v2f*)(&Bs[wn + l16][kk + 2 * hi]);
            acc = __builtin_amdgcn_wmma_f32_16x16x4_f32(
                false, a, false, b, (short)0, acc, false, false);
        }
        __syncthreads();
    }

    // epilogue: C layout m = vgpr + 8*hi, n = lane%16
    const int n = n0 + wn + l16;
    const float bv = bias[n];
    #pragma unroll
    for (int v = 0; v < 8; ++v) {
        const int m = m0 + wm + v + 8 * hi;
        const float h = acc[v] + bv;
        const float g = 0.5f * h * (1.0f + erff(h * 0.70710678118654752f));
        out[(size_t)m * WIDTH + n] = g;
    }
}

// ---------------------------------------------------------------------------
// 5) Spectral branch: thread = one width column, 4 tokens; 1024 freqs.
//    2 coalesced loads amortized over 4 sin+fma; then block RMS and accumulate.
// ---------------------------------------------------------------------------
__global__ __launch_bounds__(512) void spectral_kernel(const float* __restrict__ xn,
                                                       const float* __restrict__ freq,
                                                       const float* __restrict__ thr,
                                                       float* __restrict__ out) {
    const int w  = threadIdx.x;          // 0..511
    const int nb = blockIdx.x * 4;       // token base

    float xs[4];
    #pragma unroll
    for (int j = 0; j < 4; ++j)
        xs[j] = xn[(size_t)(nb + j) * WIDTH + w] * SPECTRAL_SCALE;

    float acc[4] = {0.0f, 0.0f, 0.0f, 0.0f};
    const float* pf = freq + w;
    const float* pt = thr + w;
    #pragma unroll 2
    for (int c = 0; c < FTOT; ++c) {
        const float fr = pf[(size_t)c * WIDTH];
        const float tw = pt[(size_t)c * WIDTH];
        #pragma unroll
        for (int j = 0; j < 4; ++j)
            acc[j] = fmaf(__sinf(xs[j] * fr), tw, acc[j]);
    }

    // RMS over width for each of the 4 tokens
    float ss[4];
    #pragma unroll
    for (int j = 0; j < 4; ++j) ss[j] = acc[j] * acc[j];
    #pragma unroll
    for (int o = 16; o > 0; o >>= 1) {
        #pragma unroll
        for (int j = 0; j < 4; ++j) ss[j] += __shfl_xor(ss[j], o, 32);
    }
    __shared__ float red[16][4];
    __shared__ float rinv[4];
    const int lane = w & 31, wave = w >> 5;
    if (lane == 0) {
        #pragma unroll
        for (int j = 0; j < 4; ++j) red[wave][j] = ss[j];
    }
    __syncthreads();
    if (w < 4) {
        float t = 0.0f;
        #pragma unroll
        for (int k = 0; k < 16; ++k) t += red[k][w];
        rinv[w] = rsqrtf(t * (1.0f / WIDTH) + EPS_RMS);
    }
    __syncthreads();

    float* po = out + (size_t)nb * WIDTH + w;
    #pragma unroll
    for (int j = 0; j < 4; ++j)
        po[(size_t)j * WIDTH] += RHO * acc[j] * rinv[j];
}

// ---------------------------------------------------------------------------
// launch
// ---------------------------------------------------------------------------
extern "C" void kernel_launch(void* const* d_in, const int* in_sizes, int n_in,
                              void* d_out, int out_size, void* d_ws, size_t ws_size,
                              hipStream_t stream) {
    const float* x     = (const float*)d_in[0];
    const float* freq  = (const float*)d_in[1];
    const float* alpha = (const float*)d_in[2];
    const float* gate  = (const float*)d_in[3];
    const float* gamma = (const float*)d_in[4];
    const float* beta  = (const float*)d_in[5];
    const float* wloc  = (const float*)d_in[6];
    const float* bloc  = (const float*)d_in[7];
    float* out = (float*)d_out;

    char* ws = (char*)d_ws;
    float*    xn   = (float*)ws;                                       // 8 MB
    float*    thr  = (float*)(ws + (size_t)NTOK * WIDTH * 4);          // 2 MB
    unsigned* amax = (unsigned*)(ws + (size_t)NTOK * WIDTH * 4
                                    + (size_t)FTOT * WIDTH * 4);       // 4 B

    absmax_init_kernel<<<1, 1, 0, stream>>>(amax);
    absmax_kernel<<<128, 256, 0, stream>>>(freq, amax, FTOT * WIDTH);
    ln_kernel<<<NTOK / 8, 256, 0, stream>>>(x, gamma, beta, xn);
    thr_kernel<<<(FTOT * WIDTH) / 256, 256, 0, stream>>>(freq, alpha, gate, amax, thr);

    dim3 gg(NTOK / 32, WIDTH / 64);
    gemm_gelu_kernel<<<gg, 256, 0, stream>>>(xn, wloc, bloc, out);

    spectral_kernel<<<NTOK / 4, 512, 0, stream>>>(xn, freq, thr, out);
}